// RBFKernelLayer_65481071399933
// MI455X (gfx1250) — compile-verified
//
#include <hip/hip_runtime.h>
#include <math.h>

// RBF layer: out[b,c] = exp(-GAMMA * (||x_b||^2 - 2 x_b.c_c + ||c_c||^2))
// f32 WMMA GEMM core (V_WMMA_F32_16X16X4_F32), double-buffered LDS staged via
// CDNA5 async DMA (global_load_async_to_lds_b128 / s_wait_asynccnt).

#define BATCH 16384
#define NCENT 4096
#define DIM   512
#define GAMMA 1.0f

#define BM 128
#define BN 256
#define BK 16
#define LDT 20                          // padded LDS row stride (floats): 80B, 16B-aligned
#define BUF_FLOATS (BM * LDT + BN * LDT) // 7680 floats = 30720 B per buffer
#define BUF_BYTES  (BUF_FLOATS * 4)

typedef __attribute__((ext_vector_type(2))) float v2f;
typedef __attribute__((ext_vector_type(8))) float v8f;

// ---------- row squared-norm kernel: one wave32 per row ----------
__global__ __launch_bounds__(256)
void rbf_rownorm_kernel(const float* __restrict__ mat, float* __restrict__ out, int rows) {
  const int lane = threadIdx.x & 31;
  const int wid  = threadIdx.x >> 5;
  const int row  = blockIdx.x * 8 + wid;
  if (row >= rows) return;
  const float4* p = (const float4*)(mat + (size_t)row * DIM);
  float s = 0.f;
#pragma unroll
  for (int i = 0; i < (DIM / 4) / 32; ++i) {
    float4 v = p[lane + i * 32];
    s = fmaf(v.x, v.x, s); s = fmaf(v.y, v.y, s);
    s = fmaf(v.z, v.z, s); s = fmaf(v.w, v.w, s);
  }
#pragma unroll
  for (int off = 16; off; off >>= 1) s += __shfl_xor(s, off, 32);
  if (lane == 0) out[row] = s;
}

// ---------- fused WMMA GEMM + RBF epilogue ----------
// block: 256 threads = 8 waves (2 along M x 4 along N); block tile 128x256,
// wave tile 64x64 = 4x4 subtiles of 16x16 -> 16 v8f accumulators.
__global__ __launch_bounds__(256)
void rbf_wmma_kernel(const float* __restrict__ X, const float* __restrict__ Cn,
                     const float* __restrict__ xsq, const float* __restrict__ csq,
                     float* __restrict__ out) {
  __shared__ float smem[2 * BUF_FLOATS];   // double-buffered A|B tiles

  const int tid  = threadIdx.x;
  const int lane = tid & 31;
  const int wid  = tid >> 5;
  const int wm   = wid & 1;                // 2 waves along M: 64 rows each
  const int wn   = wid >> 1;               // 4 waves along N: 64 cols each
  const int mBlock = blockIdx.y * BM;
  const int nBlock = blockIdx.x * BN;
  const int lh  = lane >> 4;               // half-wave 0/1
  const int l15 = lane & 15;

  // ---- per-thread async-DMA slot precompute ----
  // A tile: 128 rows x 16 cols f32 = 512 float4 slots (this thread: tid, tid+256)
  // B tile: 256 rows x 16 cols f32 = 1024 float4 slots (this thread: tid + j*256)
  const uint32_t smemBase = (uint32_t)(uintptr_t)smem;  // generic LDS ptr: low 32 = LDS offset

  uint32_t gA[2], lA[2];
#pragma unroll
  for (int i = 0; i < 2; ++i) {
    const int s = tid + i * 256, r = s >> 2, c4 = (s & 3) << 2;
    gA[i] = (uint32_t)(((mBlock + r) * DIM + c4) * 4);   // byte offset into X (k added later)
    lA[i] = (uint32_t)((r * LDT + c4) * 4);              // byte offset within buffer
  }
  uint32_t gB[4], lB[4];
#pragma unroll
  for (int j = 0; j < 4; ++j) {
    const int s = tid + j * 256, r = s >> 2, c4 = (s & 3) << 2;
    gB[j] = (uint32_t)(((nBlock + r) * DIM + c4) * 4);
    lB[j] = (uint32_t)((BM * LDT + r * LDT + c4) * 4);
  }

  v8f acc[4][4];
#pragma unroll
  for (int i = 0; i < 4; ++i)
#pragma unroll
    for (int j = 0; j < 4; ++j)
      acc[i][j] = (v8f){0.f, 0.f, 0.f, 0.f, 0.f, 0.f, 0.f, 0.f};

  // async-DMA a whole K-tile (A 128x16 + B 256x16) into buffer p
  auto stage = [&](int p, int k0) {
    const uint32_t kb = (uint32_t)(k0 * 4);
    const uint32_t lb = smemBase + (uint32_t)p * BUF_BYTES;
#pragma unroll
    for (int i = 0; i < 2; ++i)
      asm volatile("global_load_async_to_lds_b128 %0, %1, %2"
                   :: "v"(lb + lA[i]), "v"(gA[i] + kb), "s"(X) : "memory");
#pragma unroll
    for (int j = 0; j < 4; ++j)
      asm volatile("global_load_async_to_lds_b128 %0, %1, %2"
                   :: "v"(lb + lB[j]), "v"(gB[j] + kb), "s"(Cn) : "memory");
  };

  // prologue: fill buffer 0
  stage(0, 0);
  asm volatile("s_wait_asynccnt 0x0" ::: "memory");
  __syncthreads();

  int cur = 0;
  for (int k0 = 0; k0 < DIM; k0 += BK) {
    if (k0 + BK < DIM) stage(cur ^ 1, k0 + BK);   // prefetch next tile while computing

    const float* Acur = smem + cur * BUF_FLOATS;
    const float* Bcur = Acur + BM * LDT;
#pragma unroll
    for (int ks = 0; ks < BK / 4; ++ks) {
      const int col = (ks << 2) + (lh << 1);      // A/B frag: lanes 16-31 at K+2
      v2f a[4], b[4];
#pragma unroll
      for (int i = 0; i < 4; ++i)
        a[i] = *(const v2f*)&Acur[(wm * 64 + i * 16 + l15) * LDT + col];
#pragma unroll
      for (int j = 0; j < 4; ++j)
        b[j] = *(const v2f*)&Bcur[(wn * 64 + j * 16 + l15) * LDT + col];
#pragma unroll
      for (int i = 0; i < 4; ++i)
#pragma unroll
        for (int j = 0; j < 4; ++j)
          acc[i][j] = __builtin_amdgcn_wmma_f32_16x16x4_f32(
              false, a[i], false, b[j], (short)0, acc[i][j], false, false);
    }

    asm volatile("s_wait_asynccnt 0x0" ::: "memory");  // own wave's DMA landed
    __syncthreads();                                   // publish; safe to reuse buffers
    cur ^= 1;
  }

  // ---- epilogue: exp(-(xsq - 2*cross + csq)), C/D layout:
  // acc VGPR r -> (M = base + r + 8*half, N = base + lane&15)
#pragma unroll
  for (int i = 0; i < 4; ++i) {
    const int mBase = mBlock + wm * 64 + i * 16 + lh * 8;
#pragma unroll
    for (int j = 0; j < 4; ++j) {
      const int n  = nBlock + wn * 64 + j * 16 + l15;
      const float cs = csq[n];
#pragma unroll
      for (int r = 0; r < 8; ++r) {
        const int m  = mBase + r;
        const float d = xsq[m] - 2.0f * acc[i][j][r] + cs;
        out[(size_t)m * NCENT + n] = __expf(-GAMMA * d);
      }
    }
  }
}

extern "C" void kernel_launch(void* const* d_in, const int* in_sizes, int n_in,
                              void* d_out, int out_size, void* d_ws, size_t ws_size,
                              hipStream_t stream) {
  const float* x       = (const float*)d_in[0];   // [16384, 512]
  const float* centers = (const float*)d_in[1];   // [4096, 512]
  float* out = (float*)d_out;                     // [16384, 4096]

  float* xsq = (float*)d_ws;                      // 16384 floats
  float* csq = xsq + BATCH;                       // 4096 floats (80KB of ws total)

  rbf_rownorm_kernel<<<BATCH / 8, 256, 0, stream>>>(x, xsq, BATCH);
  rbf_rownorm_kernel<<<NCENT / 8, 256, 0, stream>>>(centers, csq, NCENT);

  dim3 grid(NCENT / BN, BATCH / BM);   // (16, 128)
  rbf_wmma_kernel<<<grid, 256, 0, stream>>>(x, centers, xsq, csq, out);
}